// minimal_gibbs_5076651344517
// MI455X (gfx1250) — compile-verified
//
#include <hip/hip_runtime.h>
#include <math.h>

// ---------------------------------------------------------------------------
// Gibbs-VAE on 262144 independent 5x5 images, GEMM-ified for CDNA5 WMMA.
//   encoder:  Y[b,0:150] = xext[b,0:28] @ A_enc[28,160]   (f32 WMMA 16x16x4)
//   decoder:  out[b,0:25] = zext[b,0:28] @ A_dec[28,32]
// A_enc / A_dec are collapsed weight matrices built on-device in a setup
// kernel (bias folded in as a constant-1 K-row).  Gibbs iterations run in LDS.
// A_enc is staged to LDS by the Tensor Data Mover (tensor_load_to_lds).
// ---------------------------------------------------------------------------

typedef __attribute__((ext_vector_type(2))) float v2f;
typedef __attribute__((ext_vector_type(8))) float v8f;
typedef __attribute__((ext_vector_type(4))) unsigned int v4u;
typedef __attribute__((ext_vector_type(4))) int v4i;
typedef __attribute__((ext_vector_type(8))) int v8i;

#if defined(__has_builtin)
#if __has_builtin(__builtin_amdgcn_tensor_load_to_lds) && \
    __has_builtin(__builtin_amdgcn_s_wait_tensorcnt)
#define HAVE_TDM 1
#endif
#endif

#define AENC_K   28
#define AENC_N   160
#define ADEC_N   32
#define WS_AENC  0
#define WS_ADEC  (AENC_K * AENC_N)              // 4480
#define WS_KEYS  (WS_ADEC + AENC_K * ADEC_N)    // 5376 (4 uints)

#define IPB 64    // images per block (4 waves x 16)
#define TPB 128   // 4 waves

// shared-memory float offsets (single 62720-byte block, A_enc at LDS offset 0)
#define S_AENC 0                    // 4480 floats
#define S_PM   4480                 // 1600: mu, later prec*mu
#define S_RS   (4480 + 1600)        // 1600: prec_pre, later prec_new
#define S_CP   (6080 + 1600)        // 6400: relu(coupling)
#define S_ZZ   (7680 + 6400)        // 1600: z
#define S_TOT  (14080 + 1600)       // 15680 floats = 62720 B

// ------------------------- threefry2x32 (JAX-compatible) -------------------
__device__ __forceinline__ void threefry2x32(unsigned k0, unsigned k1,
                                             unsigned c0, unsigned c1,
                                             unsigned* r0, unsigned* r1) {
  const unsigned ks2 = k0 ^ k1 ^ 0x1BD11BDAu;
  unsigned x0 = c0 + k0, x1 = c1 + k1;
  const unsigned rotA[4] = {13u, 15u, 26u, 6u};
  const unsigned rotB[4] = {17u, 29u, 16u, 24u};
#define TF_R(r) { x0 += x1; x1 = (x1 << (r)) | (x1 >> (32 - (r))); x1 ^= x0; }
#pragma unroll
  for (int i = 0; i < 4; i++) TF_R(rotA[i]);
  x0 += k1;  x1 += ks2 + 1u;
#pragma unroll
  for (int i = 0; i < 4; i++) TF_R(rotB[i]);
  x0 += ks2; x1 += k0 + 2u;
#pragma unroll
  for (int i = 0; i < 4; i++) TF_R(rotA[i]);
  x0 += k0;  x1 += k1 + 3u;
#pragma unroll
  for (int i = 0; i < 4; i++) TF_R(rotB[i]);
  x0 += k1;  x1 += ks2 + 4u;
#pragma unroll
  for (int i = 0; i < 4; i++) TF_R(rotA[i]);
  x0 += ks2; x1 += k0 + 5u;
#undef TF_R
  *r0 = x0; *r1 = x1;
}

// Giles (2012) single-precision erfinv
__device__ __forceinline__ float erfinv_f(float xx) {
  float w = -logf((1.0f - xx) * (1.0f + xx));
  float p;
  if (w < 5.0f) {
    w = w - 2.5f;
    p = 2.81022636e-08f;
    p = fmaf(p, w, 3.43273939e-07f);
    p = fmaf(p, w, -3.5233877e-06f);
    p = fmaf(p, w, -4.39150654e-06f);
    p = fmaf(p, w, 0.00021858087f);
    p = fmaf(p, w, -0.00125372503f);
    p = fmaf(p, w, -0.00417768164f);
    p = fmaf(p, w, 0.246640727f);
    p = fmaf(p, w, 1.50140941f);
  } else {
    w = sqrtf(w) - 3.0f;
    p = -0.000200214257f;
    p = fmaf(p, w, 0.000100950558f);
    p = fmaf(p, w, 0.00134934322f);
    p = fmaf(p, w, -0.00367342844f);
    p = fmaf(p, w, 0.00573950773f);
    p = fmaf(p, w, -0.0076224613f);
    p = fmaf(p, w, 0.00943887047f);
    p = fmaf(p, w, 1.00167406f);
    p = fmaf(p, w, 2.83297682f);
  }
  return p * xx;
}

__device__ __forceinline__ float bits_to_normal(unsigned bits) {
  // jax: 23 random mantissa bits -> [1,2) -> [0,1); then uniform(lo,1), lo=nextafter(-1,0)
  float f01 = __uint_as_float((bits >> 9) | 0x3f800000u) - 1.0f;
  const float lo = -0.99999994f;
  float u = fmaf(f01, 1.0f - lo, lo);
  return 1.41421356237f * erfinv_f(u);
}

// ------------------------------ setup kernel -------------------------------
__global__ void gibbs_setup(const float* __restrict__ enc_w, const float* __restrict__ enc_b,
                            const float* __restrict__ mu_w, const float* __restrict__ mu_b,
                            const float* __restrict__ prec_w, const float* __restrict__ prec_b,
                            const float* __restrict__ coup_w, const float* __restrict__ coup_b,
                            const float* __restrict__ dec1_w, const float* __restrict__ dec1_b,
                            const float* __restrict__ dec2_w, const float* __restrict__ dec2_b,
                            float* __restrict__ ws, float* __restrict__ out,
                            const float* __restrict__ lvf, int B) {
  const int tid = threadIdx.x;

  // A_enc[q][j]: q = input pixel (iy,ix) (q==25 -> bias row), j = output col
  for (int idx = tid; idx < AENC_K * AENC_N; idx += blockDim.x) {
    int q = idx / AENC_N, j = idx % AENC_N;
    float v = 0.0f;
    if (j < 150) {
      const float* wv; float bias; int p;
      if (j < 25)      { wv = mu_w;            bias = mu_b[0];   p = j; }
      else if (j < 50) { wv = prec_w;          bias = prec_b[0]; p = j - 25; }
      else             { int k = (j - 50) / 25;
                         wv = coup_w + k * 10; bias = coup_b[k]; p = (j - 50) % 25; }
      if (q < 25) {
        int iy = q / 5, ix = q % 5, y = p / 5, x = p % 5;
        int ky = iy - y + 1, kx = ix - x + 1;
        if (ky >= 0 && ky < 3 && kx >= 0 && kx < 3) {
          float s = 0.0f;
          for (int c = 0; c < 10; c++) s += wv[c] * enc_w[c * 9 + ky * 3 + kx];
          v = s;
        }
      } else if (q == 25) {  // bias row: 1x1 bias + conv bias routed through 1x1 weights
        float s = bias;
        for (int c = 0; c < 10; c++) s += wv[c] * enc_b[c];
        v = s;
      }
    }
    ws[WS_AENC + idx] = v;
  }

  // A_dec[q][p]: D = sum_c dec1_w[c] * dec2_w[c, iy-y+1, ix-x+1]; row 25 = per-pixel const
  for (int idx = tid; idx < AENC_K * ADEC_N; idx += blockDim.x) {
    int q = idx / ADEC_N, p = idx % ADEC_N;
    float v = 0.0f;
    if (p < 25) {
      int y = p / 5, x = p % 5;
      if (q < 25) {
        int iy = q / 5, ix = q % 5;
        int ky = iy - y + 1, kx = ix - x + 1;
        if (ky >= 0 && ky < 3 && kx >= 0 && kx < 3) {
          float s = 0.0f;
          for (int c = 0; c < 10; c++) s += dec1_w[c] * dec2_w[c * 9 + ky * 3 + kx];
          v = s;
        }
      } else if (q == 25) {
        float s = dec2_b[0];
        for (int c = 0; c < 10; c++) {
          float t = 0.0f;
          for (int ky = 0; ky < 3; ky++)
            for (int kx = 0; kx < 3; kx++) {
              int iy = y + ky - 1, ix = x + kx - 1;
              if (iy >= 0 && iy < 5 && ix >= 0 && ix < 5) t += dec2_w[c * 9 + ky * 3 + kx];
            }
          s += dec1_b[c] * t;
        }
        v = s;
      }
    }
    ws[WS_ADEC + idx] = v;
  }

  if (tid == 0) {
    // jax.random.key(42) -> (0,42); split(.,2): threefry over iota(4), pairs (0,2),(1,3)
    unsigned a0, a1, b0, b1;
    threefry2x32(0u, 42u, 0u, 2u, &a0, &a1);
    threefry2x32(0u, 42u, 1u, 3u, &b0, &b1);
    unsigned* kz = (unsigned*)(ws + WS_KEYS);
    kz[0] = a0; kz[1] = b0;   // k0
    kz[2] = a1; kz[3] = b1;   // k1
    out[(long long)175 * B] = lvf[0];  // log_var_final passthrough
  }
}

// --------------------- guarded helpers (CHK=false on full blocks) ----------
template <bool CHK>
__device__ __forceinline__ void scatter_enc(const v8f& acc, int j, int li0,
                                            long long img0, int B, float* sm,
                                            float* __restrict__ out,
                                            long long o_mu, long long o_cp) {
#pragma unroll
  for (int r = 0; r < 8; r++) {
    int li = li0 + r;
    long long img = img0 + li;
    if (CHK && img >= B) break;
    float v = acc[r];
    if (j < 25) {
      sm[S_PM + li * 25 + j] = v;                 // mu (pre)
      out[o_mu + img * 25 + j] = v;
    } else if (j < 50) {
      sm[S_RS + li * 25 + (j - 25)] = v;          // prec (pre-relu)
    } else if (j < 150) {
      float rv = fmaxf(v, 0.0f);
      int jj = j - 50;
      sm[S_CP + li * 100 + jj] = rv;              // relu(coupling)
      out[o_cp + img * 100 + jj] = rv;
    }
  }
}

template <bool CHK>
__device__ __forceinline__ void init_pass(int tid, long long img0, int B, float* sm,
                                          float* __restrict__ out, long long o_nlp,
                                          unsigned k0a, unsigned k0b, unsigned n2a) {
  for (int e = tid; e < IPB * 25; e += TPB) {
    int li = e / 25, p = e % 25;
    long long img = img0 + li;
    if (CHK && img >= B) continue;
    float mu = sm[S_PM + e];
    float pp = sm[S_RS + e];
    float prec = fmaxf(pp, 0.0f) + 1e-5f;
    out[o_nlp + img * 25 + p] = -logf(prec);
    unsigned i0 = (unsigned)(img * 25 + p);
    unsigned jj = (i0 < n2a) ? i0 : i0 - n2a;
    unsigned r0, r1;
    threefry2x32(k0a, k0b, jj, jj + n2a, &r0, &r1);
    float nz = bits_to_normal((i0 < n2a) ? r0 : r1);
    float cs = sm[S_CP + li * 100 + p]      + sm[S_CP + li * 100 + 25 + p] +
               sm[S_CP + li * 100 + 50 + p] + sm[S_CP + li * 100 + 75 + p];
    sm[S_ZZ + e] = fmaf(nz, rsqrtf(prec), mu);
    sm[S_PM + e] = prec * mu;        // prec*mu
    sm[S_RS + e] = prec + cs;        // prec_new
  }
}

template <bool CHK>
__device__ __forceinline__ void store_dec(const v8f& acc, int p, long long img_first,
                                          int B, float* __restrict__ out) {
#pragma unroll
  for (int r = 0; r < 8; r++) {
    long long img = img_first + r;
    if (CHK && img >= B) break;
    if (p < 25) out[img * 25 + p] = acc[r];
  }
}

// ------------------------------- main kernel -------------------------------
__global__ __launch_bounds__(TPB)
void gibbs_main(const float* __restrict__ x, const float* __restrict__ ws,
                float* __restrict__ out, int B) {
  __shared__ float sm[S_TOT];
  const int tid  = threadIdx.x;
  const int lane = tid & 31;
  const int wave = tid >> 5;
  const int img0 = blockIdx.x * IPB;
  const int la   = lane & 15;
  const int koff = (lane >> 4) * 2;           // K pair offset within 16x4 A fragment
  const int mrow = (lane >> 4) * 8;           // C fragment row base
  const bool full = (img0 + IPB) <= B;        // block-uniform: no bounds checks needed
  const long long o_mu  = (long long)B * 25;
  const long long o_nlp = (long long)B * 50;
  const long long o_cp  = (long long)B * 75;

  const unsigned* kz = (const unsigned*)(ws + WS_KEYS);
  const unsigned k0a = kz[0], k0b = kz[1], k1a = kz[2], k1b = kz[3];

  // hint the next block's x tile toward the caches (global_prefetch_b8)
  if (tid == 0) {
    long long nb = (long long)(img0 + IPB) * 25;
    if (nb < (long long)B * 25) __builtin_prefetch(x + nb, 0, 1);
  }

  // ---- phase 0: stage A_enc into LDS via the Tensor Data Mover ------------
#ifdef HAVE_TDM
  if (wave == 0) {
    // D# group0: count=1 | lds_addr | 57-bit global_addr | type=2 ("image")
    unsigned long long ga = (unsigned long long)(const void*)(ws + WS_AENC);
    v4u g0;
    g0[0] = 1u;                                   // count=1, user descriptor
    g0[1] = (unsigned)(S_AENC * 4);               // LDS byte address (sm starts at 0)
    g0[2] = (unsigned)(ga & 0xffffffffu);         // global_addr[31:0]
    g0[3] = (unsigned)((ga >> 32) & 0x1ffffffu)   // global_addr[56:32]
            | (2u << 30);                         // type = 2
    // D# group1: data_size=4B, 1 row of 4480 elements (17,920 B contiguous)
    v8i g1;
    g1[0] = 0x20000;                // data_size=2 (4B); no multicast/flags/padding
    g1[1] = (int)(4480u << 16);     // tensor_dim0[15:0] in [63:48]
    g1[2] = (int)(1u << 16);        // tensor_dim0 hi=0 | tensor_dim1=1
    g1[3] = (int)(4480u << 16);     // tensor_dim1 hi=0 | tile_dim0=4480
    g1[4] = 1;                      // tile_dim1=1, tile_dim2=0 (unused)
    g1[5] = 4480;                   // tensor_dim0_stride[31:0]
    g1[6] = 0;                      // stride hi | tensor_dim1_stride lo
    g1[7] = 0;
    v4i z4 = {0, 0, 0, 0};                      // groups 2/3: dims 3/4 unused
    v8i z8 = {0, 0, 0, 0, 0, 0, 0, 0};          // extra descriptor words (unused)
    __builtin_amdgcn_tensor_load_to_lds(g0, g1, z4, z4, z8, 0);
    __builtin_amdgcn_s_wait_tensorcnt(0);
  }
#else
  for (int i = tid; i < AENC_K * AENC_N; i += TPB) sm[S_AENC + i] = ws[WS_AENC + i];
#endif
  __syncthreads();

  // ---------------- phase 1: encoder GEMM (per wave: 16 images) ------------
  const int wimg = img0 + wave * 16;
  v2f av[7];
  {
    const int img = wimg + la;
    const bool ok = img < B;
    const float* xp = x + (long long)img * 25;
#pragma unroll
    for (int t = 0; t < 7; t++) {
      int q0 = 4 * t + koff, q1 = q0 + 1;
      float a0 = !ok ? 0.0f : (q0 < 25 ? xp[q0] : (q0 == 25 ? 1.0f : 0.0f));
      float a1 = !ok ? 0.0f : (q1 < 25 ? xp[q1] : (q1 == 25 ? 1.0f : 0.0f));
      v2f t2; t2.x = a0; t2.y = a1; av[t] = t2;
    }
  }

#pragma unroll
  for (int nt = 0; nt < 10; nt++) {
    v8f acc = {};
#pragma unroll
    for (int t = 0; t < 7; t++) {
      int kk = 4 * t + koff;
      v2f b;
      b.x = sm[S_AENC + kk * AENC_N + nt * 16 + la];
      b.y = sm[S_AENC + (kk + 1) * AENC_N + nt * 16 + la];
      acc = __builtin_amdgcn_wmma_f32_16x16x4_f32(false, av[t], false, b,
                                                  (short)0, acc, false, false);
    }
    const int j = nt * 16 + la;
    const int li0 = wave * 16 + mrow;
    if (full) scatter_enc<false>(acc, j, li0, img0, B, sm, out, o_mu, o_cp);
    else      scatter_enc<true>(acc, j, li0, img0, B, sm, out, o_mu, o_cp);
  }
  __syncthreads();

  // ---------------- phase 1.5: elementwise init ----------------------------
  const unsigned n2a = (unsigned)(B * 25) >> 1;       // noise0 pool half-size
  if (full) init_pass<false>(tid, img0, B, sm, out, o_nlp, k0a, k0b, n2a);
  else      init_pass<true>(tid, img0, B, sm, out, o_nlp, k0a, k0b, n2a);
  __syncthreads();

  // ---------------- phase 2: 5 Gibbs sweeps in LDS -------------------------
  const unsigned n2b = (unsigned)(5 * B * 25) >> 1;   // step-noise pool half-size
  for (int s = 0; s < 5; s++) {
    float znew[13];
#pragma unroll
    for (int t = 0; t < 13; t++) {
      int e = tid + t * TPB;
      if (e < IPB * 25) {
        int li = e / 25, p = e % 25;
        int y = p / 5, xq = p % 5;
        float num = sm[S_PM + e];
        if (y  < 4) num = fmaf(sm[S_CP + li * 100 +      p], sm[S_ZZ + li * 25 + p + 5], num);
        if (xq < 4) num = fmaf(sm[S_CP + li * 100 + 25 + p], sm[S_ZZ + li * 25 + p + 1], num);
        if (y  > 0) num = fmaf(sm[S_CP + li * 100 + 50 + p], sm[S_ZZ + li * 25 + p - 5], num);
        if (xq > 0) num = fmaf(sm[S_CP + li * 100 + 75 + p], sm[S_ZZ + li * 25 + p - 1], num);
        float pn = sm[S_RS + e];
        long long img = img0 + li;
        unsigned idx = (unsigned)(((long long)s * B + img) * 25 + p);
        unsigned jj = (idx < n2b) ? idx : idx - n2b;
        unsigned r0, r1;
        threefry2x32(k1a, k1b, jj, jj + n2b, &r0, &r1);
        float nz = bits_to_normal((idx < n2b) ? r0 : r1);
        znew[t] = fmaf(nz, rsqrtf(pn), num / pn);
      }
    }
    __syncthreads();
#pragma unroll
    for (int t = 0; t < 13; t++) {
      int e = tid + t * TPB;
      if (e < IPB * 25) sm[S_ZZ + e] = znew[t];
    }
    __syncthreads();
  }

  // ---------------- phase 3: decoder GEMM (per wave) -----------------------
  v2f zv[7];
  {
    int li = wave * 16 + la;
#pragma unroll
    for (int t = 0; t < 7; t++) {
      int q0 = 4 * t + koff, q1 = q0 + 1;
      float a0 = q0 < 25 ? sm[S_ZZ + li * 25 + q0] : (q0 == 25 ? 1.0f : 0.0f);
      float a1 = q1 < 25 ? sm[S_ZZ + li * 25 + q1] : (q1 == 25 ? 1.0f : 0.0f);
      v2f t2; t2.x = a0; t2.y = a1; zv[t] = t2;
    }
  }
#pragma unroll
  for (int nt = 0; nt < 2; nt++) {
    v8f acc = {};
#pragma unroll
    for (int t = 0; t < 7; t++) {
      int kk = 4 * t + koff;
      v2f b;
      b.x = ws[WS_ADEC + kk * ADEC_N + nt * 16 + la];
      b.y = ws[WS_ADEC + (kk + 1) * ADEC_N + nt * 16 + la];
      acc = __builtin_amdgcn_wmma_f32_16x16x4_f32(false, zv[t], false, b,
                                                  (short)0, acc, false, false);
    }
    const int p = nt * 16 + la;
    const long long img_first = img0 + wave * 16 + mrow;
    if (full) store_dec<false>(acc, p, img_first, B, out);
    else      store_dec<true>(acc, p, img_first, B, out);
  }
}

// ------------------------------- launcher ----------------------------------
extern "C" void kernel_launch(void* const* d_in, const int* in_sizes, int n_in,
                              void* d_out, int out_size, void* d_ws, size_t ws_size,
                              hipStream_t stream) {
  (void)n_in; (void)out_size; (void)ws_size;
  const float* x      = (const float*)d_in[0];
  const float* enc_w  = (const float*)d_in[1];
  const float* enc_b  = (const float*)d_in[2];
  const float* mu_w   = (const float*)d_in[3];
  const float* mu_b   = (const float*)d_in[4];
  const float* prec_w = (const float*)d_in[5];
  const float* prec_b = (const float*)d_in[6];
  const float* coup_w = (const float*)d_in[7];
  const float* coup_b = (const float*)d_in[8];
  const float* dec1_w = (const float*)d_in[9];
  const float* dec1_b = (const float*)d_in[10];
  const float* dec2_w = (const float*)d_in[11];
  const float* dec2_b = (const float*)d_in[12];
  const float* lvf    = (const float*)d_in[13];
  float* out = (float*)d_out;
  float* ws  = (float*)d_ws;
  const int B = in_sizes[0] / 25;

  gibbs_setup<<<1, 256, 0, stream>>>(enc_w, enc_b, mu_w, mu_b, prec_w, prec_b,
                                     coup_w, coup_b, dec1_w, dec1_b, dec2_w, dec2_b,
                                     ws, out, lvf, B);
  const int blocks = (B + IPB - 1) / IPB;
  gibbs_main<<<blocks, TPB, 0, stream>>>(x, ws, out, B);
}